// STDPLinear_21311627722942
// MI455X (gfx1250) — compile-verified
//
#include <hip/hip_runtime.h>
#include <hip/hip_bf16.h>

// Problem constants (match reference)
#define BATCH 128
#define IN_F  1024
#define OUT_F 1024
#define BETA      0.99f
#define THRESH    1.0f
#define RESET     0.8f
#define INV_TAU   0.05f            // 1/20
#define A_SCALE   (0.005f / 128.0f) // A_POS(=A_NEG) / B, applied once at the end

typedef float v2f __attribute__((ext_vector_type(2)));
typedef float v8f __attribute__((ext_vector_type(8)));

// ---------------------------------------------------------------------------
// Kernel 1: weighted = in_spikes @ W.T via V_WMMA_F32_16X16X4_F32, fused LIF.
// One wave (32 lanes) owns a 16(batch) x 16(out) output tile.
// A-frag (16x4 f32, 2 VGPRs): lanes 0-15 -> M=lane, K = k+{0,1};
//                             lanes 16-31 -> M=lane-16, K = k+{2,3}.
// B-frag (4x16 f32, 2 VGPRs): lanes 0-15 -> N=lane, K = k+{0,1};
//                             lanes 16-31 -> N=lane-16, K = k+{2,3}.
// C/D (16x16 f32, 8 VGPRs): VGPR r -> M = r + 8*(lane>=16), N = lane&15.
// ---------------------------------------------------------------------------
__global__ __launch_bounds__(256)
void stdp_gemm_lif_kernel(const float* __restrict__ in_spikes,
                          const float* __restrict__ W,
                          const float* __restrict__ membrane,
                          float* __restrict__ out_spike,
                          float* __restrict__ out_mem) {
    const int wave = threadIdx.x >> 5;
    const int lane = threadIdx.x & 31;
    const int tile = blockIdx.x * 8 + wave;     // 512 tiles = 8 (M) x 64 (N)
    const int m0 = (tile & 7) << 4;             // batch tile origin
    const int n0 = (tile >> 3) << 4;            // out-feature tile origin

    const int half  = lane >> 4;                // 0: K lo pair, 1: K hi pair
    const int sub   = lane & 15;
    const int khalf = half << 1;                // 0 or 2

    const float* __restrict__ arow = in_spikes + (size_t)(m0 + sub) * IN_F;
    const float* __restrict__ brow = W        + (size_t)(n0 + sub) * IN_F;

    v8f c = {};
    for (int k = 0; k < IN_F; k += 4) {
        v2f a = *(const v2f*)(arow + k + khalf);   // 8B-aligned (k%4==0)
        v2f b = *(const v2f*)(brow + k + khalf);
        c = __builtin_amdgcn_wmma_f32_16x16x4_f32(
                /*neg_a=*/false, a, /*neg_b=*/false, b,
                /*c_mod=*/(short)0, c, /*reuse_a=*/false, /*reuse_b=*/false);
    }

    // Fused LIF epilogue: mem = membrane*beta + weighted; spike; reset.
    const int n     = n0 + sub;
    const int mbase = m0 + (half << 3);
#pragma unroll
    for (int r = 0; r < 8; ++r) {
        const size_t idx = (size_t)(mbase + r) * OUT_F + n;
        float mem = membrane[idx] * BETA + c[r];
        float spk = (mem > THRESH) ? 1.0f : 0.0f;
        out_spike[idx] = spk;
        out_mem[idx]   = (spk > 0.0f) ? (mem - RESET) : mem;
    }
}

// ---------------------------------------------------------------------------
// Kernel 2: STDP weight update.
//   dp = in_spike ? 0 : delta_pre+1   (per b,i)
//   df = spike    ? 0 : delta_fire+1  (per b,o)
//   wc[b,i,o] = (df>dp)*A*exp(-(df-dp)/tau) - (df<dp)*A*exp((df-dp)/tau)
// Separated:  exp(-(df-dp)/tau) = exp(-df/tau)*exp(dp/tau), etc.
// Comparison done in exp-domain (monotonic, values in [e^-2.55, e^2.55]).
// Block = 16(out) x 16(in) tile of W; 32KB LDS holds the 4 staged exp arrays
// for all 128 batches; inner loop is pure VALU + ds reads.
// ---------------------------------------------------------------------------
__global__ __launch_bounds__(256)
void stdp_weight_kernel(const float* __restrict__ in_spikes,
                        const float* __restrict__ delta_pre,
                        const float* __restrict__ delta_fire,
                        const float* __restrict__ spike,
                        const float* __restrict__ W,
                        float* __restrict__ out_W) {
    __shared__ float s_ep_i[BATCH][16];   // exp(+dp/tau)
    __shared__ float s_en_i[BATCH][16];   // exp(-dp/tau)
    __shared__ float s_ep_o[BATCH][16];   // exp(+df/tau)
    __shared__ float s_en_o[BATCH][16];   // exp(-df/tau)

    const int i0 = blockIdx.x << 4;       // input-feature tile origin
    const int o0 = blockIdx.y << 4;       // output-feature tile origin

    // Stage exp terms for the whole batch (2048 elems/side, 8 per thread).
    for (int idx = threadIdx.x; idx < BATCH * 16; idx += 256) {
        const int b = idx >> 4;
        const int col = idx & 15;
        // presynaptic side
        {
            const size_t g = (size_t)b * IN_F + (i0 + col);
            const float dpv = (in_spikes[g] > 0.0f) ? 0.0f : (delta_pre[g] + 1.0f);
            s_ep_i[b][col] = __expf(dpv *  INV_TAU);
            s_en_i[b][col] = __expf(dpv * -INV_TAU);
        }
        // postsynaptic side
        {
            const size_t g = (size_t)b * OUT_F + (o0 + col);
            const float dfv = (spike[g] > 0.0f) ? 0.0f : (delta_fire[g] + 1.0f);
            s_ep_o[b][col] = __expf(dfv *  INV_TAU);
            s_en_o[b][col] = __expf(dfv * -INV_TAU);
        }
    }
    __syncthreads();

    const int o_l = threadIdx.x >> 4;
    const int i_l = threadIdx.x & 15;

    float acc_pot = 0.0f;
    float acc_dep = 0.0f;
#pragma unroll 8
    for (int b = 0; b < BATCH; ++b) {
        const float ep_i = s_ep_i[b][i_l];
        const float en_i = s_en_i[b][i_l];
        const float ep_o = s_ep_o[b][o_l];
        const float en_o = s_en_o[b][o_l];
        // df > dp  <=>  exp(df/tau) > exp(dp/tau)
        acc_pot += (ep_o > ep_i) ? en_o * ep_i : 0.0f;
        acc_dep += (ep_o < ep_i) ? ep_o * en_i : 0.0f;
    }

    const size_t w_idx = (size_t)(o0 + o_l) * IN_F + (i0 + i_l);
    out_W[w_idx] = W[w_idx] + (acc_pot - acc_dep) * A_SCALE;
}

// ---------------------------------------------------------------------------
// Host launcher. Inputs (setup_inputs order):
//   0: in_spikes (B,IN)  1: W (OUT,IN)  2: membrane (B,OUT)
//   3: delta_pre (B,IN)  4: delta_fire (B,OUT)
// Output tuple (spike, new_W, mem) concatenated flat in d_out.
// ---------------------------------------------------------------------------
extern "C" void kernel_launch(void* const* d_in, const int* in_sizes, int n_in,
                              void* d_out, int out_size, void* d_ws, size_t ws_size,
                              hipStream_t stream) {
    (void)in_sizes; (void)n_in; (void)out_size; (void)d_ws; (void)ws_size;

    const float* in_spikes  = (const float*)d_in[0];
    const float* W          = (const float*)d_in[1];
    const float* membrane   = (const float*)d_in[2];
    const float* delta_pre  = (const float*)d_in[3];
    const float* delta_fire = (const float*)d_in[4];

    float* out        = (float*)d_out;
    float* out_spike  = out;                                   // B*OUT
    float* out_W      = out + (size_t)BATCH * OUT_F;           // OUT*IN
    float* out_mem    = out_W + (size_t)OUT_F * IN_F;          // B*OUT

    // 512 WMMA tiles, 8 waves (256 threads) per block -> 64 blocks.
    stdp_gemm_lif_kernel<<<64, 256, 0, stream>>>(
        in_spikes, W, membrane, out_spike, out_mem);

    // 64 x 64 tiles of the 1024x1024 weight matrix.
    stdp_weight_kernel<<<dim3(IN_F / 16, OUT_F / 16), 256, 0, stream>>>(
        in_spikes, delta_pre, delta_fire, out_spike, W, out_W);
}